// WindowAttention_50714973831592
// MI455X (gfx1250) — compile-verified
//
#include <hip/hip_runtime.h>
#include <math.h>

// Problem constants (fixed by the reference's setup_inputs).
constexpr int BATCH = 2;
constexpr int NTOK  = 65536;   // 256*256 tokens
constexpr int CDIM  = 256;     // channels
constexpr int NGRP  = 512;     // Voronoi cells
constexpr int SGRP  = 128;     // tokens per cell
constexpr int NHEAD = 8;
constexpr int HDIM  = 32;      // head dim
constexpr int NCHUNK = 256;    // sort chunks of 256 tokens

typedef float v2f __attribute__((ext_vector_type(2)));
typedef float v4f __attribute__((ext_vector_type(4)));
typedef float v8f __attribute__((ext_vector_type(8)));

// D(16x16,f32) = A(16x4,f32) x B(4x16,f32) + C   — full fp32 precision on the matrix pipe.
__device__ __forceinline__ v8f wmma4(v2f a, v2f b, v8f c) {
  return __builtin_amdgcn_wmma_f32_16x16x4_f32(
      /*neg_a=*/false, a, /*neg_b=*/false, b,
      /*c_mod=*/(short)0, c, /*reuse_a=*/false, /*reuse_b=*/false);
}

// ---------------------------------------------------------------------------
// Deterministic stable counting sort of tokens by Voronoi label.
// ---------------------------------------------------------------------------

// Pass 1: per-chunk label histogram (LDS atomics; result is a pure sum -> deterministic).
__global__ __launch_bounds__(256) void sort_count(const int* __restrict__ labels,
                                                  int* __restrict__ cnt) {
  __shared__ int c[NGRP];
  const int b = blockIdx.y, chunk = blockIdx.x, tid = threadIdx.x;
  for (int i = tid; i < NGRP; i += 256) c[i] = 0;
  __syncthreads();
  const int lab = labels[(size_t)b * NTOK + chunk * 256 + tid];
  atomicAdd(&c[lab - 1], 1);
  __syncthreads();
  for (int i = tid; i < NGRP; i += 256)
    cnt[((size_t)b * NCHUNK + chunk) * NGRP + i] = c[i];
}

// Pass 2: exclusive prefix over chunks, per (batch,label).
__global__ __launch_bounds__(256) void sort_prefix(const int* __restrict__ cnt,
                                                   int* __restrict__ base) {
  const int b = blockIdx.y;
  const int g = blockIdx.x * 256 + threadIdx.x;   // 0..511
  int run = 0;
  for (int ch = 0; ch < NCHUNK; ++ch) {
    base[((size_t)b * NCHUNK + ch) * NGRP + g] = run;
    run += cnt[((size_t)b * NCHUNK + ch) * NGRP + g];
  }
}

// Pass 3: stable in-chunk rank (brute-force scan over LDS) -> order[] and pos[].
__global__ __launch_bounds__(256) void sort_assign(const int* __restrict__ labels,
                                                   const int* __restrict__ base,
                                                   int* __restrict__ order,
                                                   int* __restrict__ pos) {
  __shared__ int lab[256];
  const int b = blockIdx.y, chunk = blockIdx.x, tid = threadIdx.x;
  const int t = chunk * 256 + tid;
  lab[tid] = labels[(size_t)b * NTOK + t];
  __syncthreads();
  const int mylab = lab[tid];
  int rank = 0;
  for (int j = 0; j < tid; ++j) rank += (lab[j] == mylab);
  const int g = mylab - 1;
  const int p = g * SGRP + base[((size_t)b * NCHUNK + chunk) * NGRP + g] + rank;
  order[(size_t)b * NTOK + p] = t;
  pos[(size_t)b * NTOK + t] = p;
}

// ---------------------------------------------------------------------------
// Fused GEMM: dst[sidx[m], :] = (X[m, :] @ W^T + bias) * scale
// X: [B,N,C] rows read sequentially; W: [C,C]; store rows scattered by sidx.
// WG = 256 threads = 8 waves; tile 64(M) x 128(N); K staged in 64-chunks.
// Used for q/k/v projections (scatter to grouped order) and for the output
// projection (scatter from grouped order back to token order).
// ---------------------------------------------------------------------------
__global__ __launch_bounds__(256) void gemm_scatter(const float* __restrict__ X,
                                                    const float* __restrict__ W,
                                                    const float* __restrict__ bias,
                                                    const int* __restrict__ sidx,
                                                    float* __restrict__ dst,
                                                    float scale) {
  __shared__ float Xs[64][68];    // +4 pad: 16 consecutive rows span all 64 banks
  __shared__ float Ws[128][68];
  const int b  = blockIdx.z;
  const int m0 = blockIdx.x * 64;
  const int nh = blockIdx.y;                 // column half: [nh*128, nh*128+128)
  const int tid = threadIdx.x;
  const int lane = tid & 31, wave = tid >> 5;
  const int l15 = lane & 15, h = lane >> 4;
  const int rb = wave & 3;                   // row tile 0..3 (16 rows each)
  const int cp = wave >> 2;                  // column quad 0..1 (4 tiles of 16)

  v8f acc[4] = {v8f{}, v8f{}, v8f{}, v8f{}};

  for (int kc = 0; kc < 4; ++kc) {           // K in chunks of 64
    __syncthreads();
    for (int i = tid; i < 64 * 64; i += 256) {
      const int r = i >> 6, k = i & 63;
      Xs[r][k] = X[((size_t)b * NTOK + m0 + r) * CDIM + kc * 64 + k];
    }
    for (int i = tid; i < 128 * 64; i += 256) {
      const int r = i >> 6, k = i & 63;
      Ws[r][k] = W[((size_t)(nh * 128 + r)) * CDIM + kc * 64 + k];
    }
    __syncthreads();
    const float* xrow = &Xs[rb * 16 + l15][0];
    for (int ct = 0; ct < 4; ++ct) {
      const float* wrow = &Ws[cp * 64 + ct * 16 + l15][0];
      v8f a8 = acc[ct];
      for (int kk = 0; kk < 64; kk += 4) {
        const v2f av = *(const v2f*)&xrow[kk + 2 * h];  // A[l15][2h..2h+1]
        const v2f bv = *(const v2f*)&wrow[kk + 2 * h];  // B[2h..2h+1][l15] = W[n][k]
        a8 = wmma4(av, bv, a8);
      }
      acc[ct] = a8;
    }
  }

  // Epilogue: bias + scale, scatter 16-row tiles row-by-row via sidx.
  for (int ct = 0; ct < 4; ++ct) {
    const int n = nh * 128 + cp * 64 + ct * 16 + l15;
    const float bz = bias[n];
    const v8f a8 = acc[ct];
    #pragma unroll
    for (int r = 0; r < 8; ++r) {
      const int m = m0 + rb * 16 + r + 8 * h;       // C/D layout: M = r + 8h
      const int p = sidx[(size_t)b * NTOK + m];
      dst[((size_t)b * NTOK + p) * CDIM + n] = (a8[r] + bz) * scale;
    }
  }
}

// ---------------------------------------------------------------------------
// Per-(batch,cell,head) attention: S=softmax(Q K^T) ; O = S V.  128x128, hd=32.
// One WG of 128 threads (4 waves); wave w owns 32 query rows.
// Dynamic LDS: Q/K/V tiles (pad 36) + score matrix 128x130 (~122 KB).
// ---------------------------------------------------------------------------
__global__ __launch_bounds__(128) void attn_kernel(const float* __restrict__ Qg,
                                                   const float* __restrict__ Kg,
                                                   const float* __restrict__ Vg,
                                                   float* __restrict__ og) {
  extern __shared__ float smem[];
  float (*Qs)[36]  = (float(*)[36])smem;
  float (*Ks)[36]  = (float(*)[36])(smem + 128 * 36);
  float (*Vs)[36]  = (float(*)[36])(smem + 2 * 128 * 36);
  float (*Ps)[130] = (float(*)[130])(smem + 3 * 128 * 36);

  const int bg = blockIdx.x;                  // b*NGRP + g
  const int hh = blockIdx.y;                  // head
  const int b = bg >> 9, g = bg & (NGRP - 1);
  const int tid = threadIdx.x;
  const int lane = tid & 31, wave = tid >> 5;
  const int l15 = lane & 15, h = lane >> 4;
  const size_t rowbase = (size_t)b * NTOK + (size_t)g * SGRP;

  // Cooperative load of the group's Q/K/V head slices (float4 segments).
  for (int i = tid * 4; i < SGRP * HDIM; i += 128 * 4) {
    const int r = i >> 5, c = i & 31;
    const size_t src = (rowbase + r) * CDIM + hh * HDIM + c;
    *(v4f*)&Qs[r][c] = *(const v4f*)&Qg[src];
    *(v4f*)&Ks[r][c] = *(const v4f*)&Kg[src];
    *(v4f*)&Vs[r][c] = *(const v4f*)&Vg[src];
  }
  __syncthreads();

  // Phase 1: scores = Q K^T  (scale already folded into Q at projection).
  v8f acc[2][8];
  #pragma unroll
  for (int rt = 0; rt < 2; ++rt)
    #pragma unroll
    for (int ct = 0; ct < 8; ++ct) acc[rt][ct] = v8f{};

  for (int kk = 0; kk < HDIM; kk += 4) {
    const v2f a0 = *(const v2f*)&Qs[wave * 32 + l15][kk + 2 * h];
    const v2f a1 = *(const v2f*)&Qs[wave * 32 + 16 + l15][kk + 2 * h];
    #pragma unroll
    for (int ct = 0; ct < 8; ++ct) {
      const v2f bv = *(const v2f*)&Ks[ct * 16 + l15][kk + 2 * h];  // B[K][t]=K[t][k]
      acc[0][ct] = wmma4(a0, bv, acc[0][ct]);
      acc[1][ct] = wmma4(a1, bv, acc[1][ct]);
    }
  }
  #pragma unroll
  for (int rt = 0; rt < 2; ++rt)
    #pragma unroll
    for (int ct = 0; ct < 8; ++ct) {
      const v8f a8 = acc[rt][ct];
      #pragma unroll
      for (int r = 0; r < 8; ++r)
        Ps[wave * 32 + rt * 16 + r + 8 * h][ct * 16 + l15] = a8[r];
    }
  __syncthreads();

  // Row softmax: thread i owns row i (stride 130 -> conflict-free in a wave).
  {
    float* row = &Ps[tid][0];
    float mx = -INFINITY;
    for (int t = 0; t < SGRP; ++t) mx = fmaxf(mx, row[t]);
    float sum = 0.f;
    for (int t = 0; t < SGRP; ++t) { const float e = __expf(row[t] - mx); row[t] = e; sum += e; }
    const float inv = 1.0f / sum;
    for (int t = 0; t < SGRP; ++t) row[t] *= inv;
  }
  __syncthreads();

  // Phase 2: O = P V  (32 rows per wave, 2x2 tiles of 16x16, K = 128).
  v8f oacc[2][2];
  #pragma unroll
  for (int rt = 0; rt < 2; ++rt)
    #pragma unroll
    for (int ct = 0; ct < 2; ++ct) oacc[rt][ct] = v8f{};

  for (int kk = 0; kk < SGRP; kk += 4) {
    const v2f a0 = *(const v2f*)&Ps[wave * 32 + l15][kk + 2 * h];
    const v2f a1 = *(const v2f*)&Ps[wave * 32 + 16 + l15][kk + 2 * h];
    #pragma unroll
    for (int ct = 0; ct < 2; ++ct) {
      v2f bv;
      bv.x = Vs[kk + 2 * h][ct * 16 + l15];       // B[K][c] = V[t][c]
      bv.y = Vs[kk + 2 * h + 1][ct * 16 + l15];
      oacc[0][ct] = wmma4(a0, bv, oacc[0][ct]);
      oacc[1][ct] = wmma4(a1, bv, oacc[1][ct]);
    }
  }
  #pragma unroll
  for (int rt = 0; rt < 2; ++rt)
    #pragma unroll
    for (int ct = 0; ct < 2; ++ct) {
      const v8f a8 = oacc[rt][ct];
      #pragma unroll
      for (int r = 0; r < 8; ++r) {
        const int s = wave * 32 + rt * 16 + r + 8 * h;
        og[(rowbase + s) * CDIM + hh * HDIM + ct * 16 + l15] = a8[r];
      }
    }
}

// ---------------------------------------------------------------------------
extern "C" void kernel_launch(void* const* d_in, const int* in_sizes, int n_in,
                              void* d_out, int out_size, void* d_ws, size_t ws_size,
                              hipStream_t stream) {
  const float* xq = (const float*)d_in[0];
  const float* xk = (const float*)d_in[1];
  const float* xv = (const float*)d_in[2];
  const float* Wq = (const float*)d_in[3];
  const float* bq = (const float*)d_in[4];
  const float* Wp = (const float*)d_in[5];
  const float* bp = (const float*)d_in[6];
  const int*   vor = (const int*)d_in[7];   // Voronoi labels, [B,256,256] int32
  float* out = (float*)d_out;

  // Workspace carve-out (~540 MB): sort arrays + grouped Q/K/V/O (fp32).
  char* ws = (char*)d_ws;
  size_t o = 0;
  auto take = [&](size_t bytes) -> char* {
    char* p = ws + o;
    o = (o + bytes + 255) & ~(size_t)255;
    return p;
  };
  int* order = (int*)take((size_t)BATCH * NTOK * sizeof(int));
  int* pos   = (int*)take((size_t)BATCH * NTOK * sizeof(int));
  int* cnt   = (int*)take((size_t)BATCH * NCHUNK * NGRP * sizeof(int));
  int* basep = (int*)take((size_t)BATCH * NCHUNK * NGRP * sizeof(int));
  const size_t matbytes = (size_t)BATCH * NTOK * CDIM * sizeof(float);
  float* Qg = (float*)take(matbytes);
  float* Kg = (float*)take(matbytes);
  float* Vg = (float*)take(matbytes);
  float* Og = (float*)take(matbytes);

  // 1) Deterministic stable counting sort by Voronoi label.
  sort_count <<<dim3(NCHUNK, BATCH), 256, 0, stream>>>(vor, cnt);
  sort_prefix<<<dim3(NGRP / 256, BATCH), 256, 0, stream>>>(cnt, basep);
  sort_assign<<<dim3(NCHUNK, BATCH), 256, 0, stream>>>(vor, basep, order, pos);

  // 2) Projections with shared Wq/bq (reference quirk), scattered to group order.
  //    Q additionally carries the attention scale hd^-0.5.
  const float qscale = 0.17677669529663687f;  // 1/sqrt(32)
  const dim3 ggrid(NTOK / 64, 2, BATCH);
  gemm_scatter<<<ggrid, 256, 0, stream>>>(xq, Wq, bq, pos, Qg, qscale);
  gemm_scatter<<<ggrid, 256, 0, stream>>>(xk, Wq, bq, pos, Kg, 1.0f);
  gemm_scatter<<<ggrid, 256, 0, stream>>>(xv, Wq, bq, pos, Vg, 1.0f);

  // 3) Per-cell, per-head attention on contiguous grouped tiles.
  const size_t attn_lds = (size_t)(3 * 128 * 36 + 128 * 130) * sizeof(float);
  attn_kernel<<<dim3(BATCH * NGRP, NHEAD), 128, attn_lds, stream>>>(Qg, Kg, Vg, Og);

  // 4) Output projection; rows read in grouped order, scattered back by `order`.
  gemm_scatter<<<ggrid, 256, 0, stream>>>(Og, Wp, bp, order, out, 1.0f);
}